// Sinkhorn_29394756174573
// MI455X (gfx1250) — compile-verified
//
#include <hip/hip_runtime.h>

// ---------------------------------------------------------------------------
// Sinkhorn forward_log, b=64, n=1024, tau=1.0, 10 alternating LSE iterations.
// log_s is always s - R_i - C_j, so each iteration only rewrites R or C:
//   row  iter: R_i = lse_j(s_ij - C_j)
//   col  iter: C_j = lse_i(s_ij - R_i)
// Output: exp(s - R - C).
// Batch-blocked (32 matrices = 128MB < 192MB L2) so s stays L2-resident
// across all 11 passes of a block.
// ---------------------------------------------------------------------------

#define SK_B      64
#define SK_N      1024
#define SK_BBLK   32          // matrices per L2-resident block

typedef __attribute__((ext_vector_type(2))) float v2f;
typedef __attribute__((ext_vector_type(8))) float v8f;
typedef __attribute__((ext_vector_type(4))) int   v4i;

#if defined(__has_builtin)
#  if __has_builtin(__builtin_amdgcn_global_load_async_to_lds_b128)
#    define SK_HAVE_ASYNC 1
#  endif
#  if __has_builtin(__builtin_amdgcn_s_wait_asynccnt)
#    define SK_HAVE_WAIT_ASYNC 1
#  endif
#endif
#ifndef SK_HAVE_ASYNC
#  define SK_HAVE_ASYNC 0
#endif
#ifndef SK_HAVE_WAIT_ASYNC
#  define SK_HAVE_WAIT_ASYNC 0
#endif

// Wave32 sum-reduction via V_WMMA_F32_16X16X4_F32 with a ones B-matrix.
// A layout (16x4 f32): lane L holds A[L%16][2*(L/16)+v] in vgpr v.
// With a = {p, 0}: A[r][:] = [p_r, 0, p_{r+16}, 0]; B = ones =>
// D[r][n] = p_r + p_{r+16} for every n.  D layout: vgpr v, lane n<16 ->
// D[v][n]; lane n>=16 -> D[v+8][n-16].  Sum of the 8 D vgprs on lane n plus
// the same on lane n^16 equals the full 32-lane sum (identical on all lanes).
__device__ __forceinline__ float wave_sum_wmma(float p) {
    v2f a;  a.x = p;    a.y = 0.0f;
    v2f bo; bo.x = 1.0f; bo.y = 1.0f;
    v8f c = {};
    v8f d = __builtin_amdgcn_wmma_f32_16x16x4_f32(
        /*neg_a=*/false, a, /*neg_b=*/false, bo,
        /*c_mod=*/(short)0, c, /*reuse_a=*/false, /*reuse_b=*/false);
    float t = d[0] + d[1] + d[2] + d[3] + d[4] + d[5] + d[6] + d[7];
    return t + __shfl_xor(t, 16);
}

__device__ __forceinline__ float wave_max(float m) {
    #pragma unroll
    for (int off = 16; off >= 1; off >>= 1)
        m = fmaxf(m, __shfl_xor(m, off));
    return m;
}

// ---------------------------------------------------------------------------
// Row pass: R[b][i] = lse_j(s[b][i][j] - C[b][j]).
// 256 threads = 8 waves; each wave owns one row (full 1024 cols in registers:
// 8 float4 per lane).  C[b][*] (4KB) staged in LDS, via async-to-LDS when the
// toolchain exposes the builtin.
// ---------------------------------------------------------------------------
__global__ __launch_bounds__(256) void sk_rowpass(const float* __restrict__ s,
                                                  const float* __restrict__ C,
                                                  float* __restrict__ R,
                                                  int use_c, int bofs) {
    __shared__ float cTile[SK_N];
    const int tid  = threadIdx.x;
    const int wave = tid >> 5;
    const int lane = tid & 31;

    const long long row0 = (long long)blockIdx.x * 8;     // 8 rows per WG
    const int b    = bofs + (int)(row0 >> 10);            // SK_N == 1024
    const int irow = (int)(row0 & 1023) + wave;

    if (use_c) {
#if SK_HAVE_ASYNC
        __builtin_amdgcn_global_load_async_to_lds_b128(
            (__attribute__((address_space(1))) v4i*)
                (C + ((long long)b << 10) + tid * 4),
            (__attribute__((address_space(3))) v4i*)(&cTile[tid * 4]),
            /*offset=*/0, /*cpol=*/0);
#if SK_HAVE_WAIT_ASYNC
        __builtin_amdgcn_s_wait_asynccnt(0);
#else
        asm volatile("s_wait_asynccnt 0" ::: "memory");
#endif
#else
        ((float4*)cTile)[tid] =
            ((const float4*)(C + ((long long)b << 10)))[tid];
#endif
    } else {
        float4 z; z.x = 0.f; z.y = 0.f; z.z = 0.f; z.w = 0.f;
        ((float4*)cTile)[tid] = z;
    }
    __syncthreads();

    const float* srow = s + ((long long)b << 20) + ((long long)irow << 10);

    float4 v[8];
    #pragma unroll
    for (int k = 0; k < 8; ++k) {
        float4 x = ((const float4*)srow)[k * 32 + lane];
        float4 c = ((const float4*)cTile)[k * 32 + lane];
        v[k].x = x.x - c.x;
        v[k].y = x.y - c.y;
        v[k].z = x.z - c.z;
        v[k].w = x.w - c.w;
    }

    float m = -__builtin_inff();
    #pragma unroll
    for (int k = 0; k < 8; ++k) {
        m = fmaxf(m, fmaxf(fmaxf(v[k].x, v[k].y), fmaxf(v[k].z, v[k].w)));
    }
    m = wave_max(m);

    float p = 0.0f;
    #pragma unroll
    for (int k = 0; k < 8; ++k) {
        p += __expf(v[k].x - m);
        p += __expf(v[k].y - m);
        p += __expf(v[k].z - m);
        p += __expf(v[k].w - m);
    }

    float total = wave_sum_wmma(p);           // v_wmma_f32_16x16x4_f32

    if (lane == 0)
        R[((long long)b << 10) + irow] = m + __logf(total);
}

// ---------------------------------------------------------------------------
// Column pass, phase 1: per-256-row-chunk online LSE down each column.
// grid = (N/256, SK_BBLK, 4); thread t owns column j, streams 256 rows
// (coalesced 1KB/row across the WG).  4 independent accumulator pairs break
// the exp dependency chain.
// ---------------------------------------------------------------------------
__global__ __launch_bounds__(256) void sk_colpass_partial(
        const float* __restrict__ s, const float* __restrict__ R,
        float* __restrict__ Pm, float* __restrict__ Ps, int bofs) {
    __shared__ float rTile[256];
    const int j  = blockIdx.x * 256 + threadIdx.x;
    const int b  = bofs + blockIdx.y;
    const int i0 = blockIdx.z * 256;

    rTile[threadIdx.x] = R[((long long)b << 10) + i0 + threadIdx.x];
    __syncthreads();

    const float* base = s + ((long long)b << 20) + ((long long)i0 << 10) + j;

    float m0 = -__builtin_inff(), m1 = m0, m2 = m0, m3 = m0;
    float s0 = 0.f, s1 = 0.f, s2 = 0.f, s3 = 0.f;

    #pragma unroll 4
    for (int ii = 0; ii < 256; ii += 4) {
        float x0 = base[(long long)(ii + 0) << 10] - rTile[ii + 0];
        float x1 = base[(long long)(ii + 1) << 10] - rTile[ii + 1];
        float x2 = base[(long long)(ii + 2) << 10] - rTile[ii + 2];
        float x3 = base[(long long)(ii + 3) << 10] - rTile[ii + 3];
        float n0 = fmaxf(m0, x0); s0 = s0 * __expf(m0 - n0) + __expf(x0 - n0); m0 = n0;
        float n1 = fmaxf(m1, x1); s1 = s1 * __expf(m1 - n1) + __expf(x1 - n1); m1 = n1;
        float n2 = fmaxf(m2, x2); s2 = s2 * __expf(m2 - n2) + __expf(x2 - n2); m2 = n2;
        float n3 = fmaxf(m3, x3); s3 = s3 * __expf(m3 - n3) + __expf(x3 - n3); m3 = n3;
    }
    // merge the 4 accumulator pairs
    float M = fmaxf(fmaxf(m0, m1), fmaxf(m2, m3));
    float S = s0 * __expf(m0 - M) + s1 * __expf(m1 - M)
            + s2 * __expf(m2 - M) + s3 * __expf(m3 - M);

    const long long idx = (((long long)b * 4 + blockIdx.z) << 10) + j;
    Pm[idx] = M;
    Ps[idx] = S;
}

// Column pass, phase 2: combine the 4 row-chunk partials -> C[b][j].
__global__ __launch_bounds__(256) void sk_colpass_combine(
        const float* __restrict__ Pm, const float* __restrict__ Ps,
        float* __restrict__ C, int bofs) {
    const long long t  = (long long)blockIdx.x * 256 + threadIdx.x; // SK_BBLK*N
    const long long b  = bofs + (t >> 10);
    const long long j  = t & 1023;

    float M = -__builtin_inff();
    float mv[4];
    #pragma unroll
    for (int c = 0; c < 4; ++c) {
        mv[c] = Pm[((b * 4 + c) << 10) + j];
        M = fmaxf(M, mv[c]);
    }
    float S = 0.f;
    #pragma unroll
    for (int c = 0; c < 4; ++c)
        S += Ps[((b * 4 + c) << 10) + j] * __expf(mv[c] - M);

    C[(b << 10) + j] = M + __logf(S);
}

// ---------------------------------------------------------------------------
// Final: out = exp(s - R_i - C_j), float4 per thread.
// ---------------------------------------------------------------------------
__global__ __launch_bounds__(256) void sk_final(const float* __restrict__ s,
                                                const float* __restrict__ R,
                                                const float* __restrict__ C,
                                                float* __restrict__ out,
                                                int bofs) {
    const long long t4  = (((long long)bofs) << 18)
                        + (long long)blockIdx.x * 256 + threadIdx.x;
    const long long row = t4 >> 8;          // b*1024 + i
    const long long b   = row >> 10;
    const long long j4  = t4 & 255;

    float4 x = ((const float4*)s)[t4];
    float  r = R[row];
    float4 c = ((const float4*)C)[(b << 8) + j4];

    float4 o;
    o.x = __expf(x.x - r - c.x);
    o.y = __expf(x.y - r - c.y);
    o.z = __expf(x.z - r - c.z);
    o.w = __expf(x.w - r - c.w);
    ((float4*)out)[t4] = o;
}

// ---------------------------------------------------------------------------
extern "C" void kernel_launch(void* const* d_in, const int* in_sizes, int n_in,
                              void* d_out, int out_size, void* d_ws, size_t ws_size,
                              hipStream_t stream) {
    (void)in_sizes; (void)n_in; (void)out_size; (void)ws_size;
    const float* s   = (const float*)d_in[0];
    float*       out = (float*)d_out;

    float* R  = (float*)d_ws;                 //  64K floats
    float* C  = R  + SK_B * SK_N;             //  64K floats
    float* Pm = C  + SK_B * SK_N;             // 256K floats
    float* Ps = Pm + SK_B * 4 * SK_N;         // 256K floats

    // Process 32 matrices (128MB, L2-resident) through all 11 passes, twice.
    for (int blk = 0; blk < SK_B / SK_BBLK; ++blk) {
        const int bofs = blk * SK_BBLK;
        for (int it = 0; it < 5; ++it) {
            sk_rowpass<<<SK_BBLK * SK_N / 8, 256, 0, stream>>>(
                s, C, R, it > 0 ? 1 : 0, bofs);
            sk_colpass_partial<<<dim3(SK_N / 256, SK_BBLK, 4), 256, 0, stream>>>(
                s, R, Pm, Ps, bofs);
            sk_colpass_combine<<<SK_BBLK * SK_N / 256, 256, 0, stream>>>(
                Pm, Ps, C, bofs);
        }
        sk_final<<<SK_BBLK * SK_N * SK_N / (4 * 256), 256, 0, stream>>>(
            s, R, C, out, bofs);
    }
}